// First_ODENetwork_58669253263708
// MI455X (gfx1250) — compile-verified
//
#include <hip/hip_runtime.h>

// ---------------------------------------------------------------------------
// GRU-ODE network for MI455X (gfx1250, wave32, WMMA).
//   B=256, T=512, IN=128, XH=256, G=256, H=256, DELTA_T=1.0
// Kernel 1 (xmodel): fused Linear->ReLU->Linear over 131072 rows, WMMA f32.
// Kernel 2 (scan):  whole T=512 recurrence in ONE launch; each block owns 16
//                   batch rows of h in LDS (rows are independent across the
//                   scan), waves own 16-column tiles; all GEMMs via
//                   v_wmma_f32_16x16x4_f32 in full fp32 precision.
// ---------------------------------------------------------------------------

typedef __attribute__((ext_vector_type(2))) float v2f;
typedef __attribute__((ext_vector_type(8))) float v8f;

#define B_   256
#define T_   512
#define IN_  128
#define XH_  256
#define G_   256
#define H_   256
#define LDSP 274   // padded LDS row stride in floats (274*4B; 274%64=18 -> 16 rows hit distinct banks)

__device__ __forceinline__ v8f wmma_f32_k4(v2f a, v2f b, v8f c) {
  // D = A(16x4,f32) * B(4x16,f32) + C(16x16,f32)
  return __builtin_amdgcn_wmma_f32_16x16x4_f32(
      /*neg_a=*/false, a, /*neg_b=*/false, b,
      /*c_mod=*/(short)0, c, /*reuse_a=*/false, /*reuse_b=*/false);
}

__device__ __forceinline__ float sigm(float x) { return 1.0f / (1.0f + __expf(-x)); }

// ---------------------------------------------------------------------------
// Kernel 1: HH = (relu(H @ wx1^T + bx1)) @ wx2^T + bx2      [B*T, G]
// Block: 512 threads (16 waves). Each block: 16 rows. Wave w: cols [16w,16w+16).
// ---------------------------------------------------------------------------
__global__ void __launch_bounds__(512, 1)
xmodel_kernel(const float* __restrict__ Hin,
              const float* __restrict__ wx1, const float* __restrict__ bx1,
              const float* __restrict__ wx2, const float* __restrict__ bx2,
              float* __restrict__ HHout)
{
  const int row0 = blockIdx.x * 16;          // row tile of the 131072 rows
  const int wave = threadIdx.x >> 5;         // 0..15
  const int lane = threadIdx.x & 31;
  const int half = lane >> 4;                // 0: K 0..1 / M 0..7, 1: K 2..3 / M 8..15
  const int l15  = lane & 15;
  const int n0   = wave * 16;

  __shared__ float x1[16][LDSP];             // relu(layer1) tile, 16 x 256 (padded)

  // ---- GEMM1: A = Hin rows, B = wx1^T fragments, K = 128 ----
  v8f acc = {};
  {
    const float* arow = Hin + (size_t)(row0 + l15) * IN_ + 2 * half;
    const float* brow = wx1 + (size_t)(n0 + l15) * IN_ + 2 * half;
#pragma unroll 8
    for (int k = 0; k < IN_; k += 4) {
      v2f a = *(const v2f*)(arow + k);
      v2f b = *(const v2f*)(brow + k);
      acc = wmma_f32_k4(a, b, acc);
    }
  }
  // bias + ReLU epilogue into LDS (C/D layout: M = v + 8*half, N = l15)
  {
    const float bias = bx1[n0 + l15];
#pragma unroll
    for (int v = 0; v < 8; ++v) {
      const int m = v + half * 8;
      x1[m][n0 + l15] = fmaxf(acc[v] + bias, 0.0f);
    }
  }
  __syncthreads();

  // ---- GEMM2: A = x1 (LDS), B = wx2^T fragments, K = 256 ----
  v8f acc2 = {};
  {
    const float* brow = wx2 + (size_t)(n0 + l15) * XH_ + 2 * half;
#pragma unroll 8
    for (int k = 0; k < XH_; k += 4) {
      v2f a = *(const v2f*)&x1[l15][k + 2 * half];
      v2f b = *(const v2f*)(brow + k);
      acc2 = wmma_f32_k4(a, b, acc2);
    }
  }
  {
    const float bias = bx2[n0 + l15];
#pragma unroll
    for (int v = 0; v < 8; ++v) {
      const int m = v + half * 8;
      HHout[(size_t)(row0 + m) * G_ + n0 + l15] = acc2[v] + bias;
    }
  }
}

// ---------------------------------------------------------------------------
// Kernel 2: full scan.  Grid = 16 blocks (16 batch rows each), 512 threads.
// h tile lives in LDS for all 512 steps; weights stream from L2 as WMMA
// B-fragments.  Euler counts replayed from times[0] as uniform scalar code.
// ---------------------------------------------------------------------------
__global__ void __launch_bounds__(512, 1)
scan_kernel(const float* __restrict__ HHin, const int* __restrict__ times,
            const float* __restrict__ w_hr, const float* __restrict__ w_hz,
            const float* __restrict__ w_hh,
            const float* __restrict__ w_ih, const float* __restrict__ w_hh2,
            const float* __restrict__ b_ih, const float* __restrict__ b_hh,
            float* __restrict__ out)
{
  const int m0   = blockIdx.x * 16;          // batch-row tile
  const int wave = threadIdx.x >> 5;
  const int lane = threadIdx.x & 31;
  const int half = lane >> 4;
  const int l15  = lane & 15;
  const int n0   = wave * 16;
  const int nc   = n0 + l15;                 // this lane's output column

  __shared__ float hs [16][LDSP];            // hidden state tile
  __shared__ float rhs[16][LDSP];            // r*h tile (euler)
  __shared__ float xs [16][LDSP];            // x_t tile

  // h0 = 0
  for (int i = threadIdx.x; i < 16 * H_; i += 512)
    hs[i >> 8][i & 255] = 0.0f;
  __syncthreads();

  float cur = (float)times[0];               // times[0][0]

  for (int t = 0; t < T_; ++t) {
    // -- Euler step count (uniform across block; matches host replay) --
    const float tt = (float)times[t];
    int kcnt = 0;
    while (cur < tt - 0.001f) { cur += 1.0f; ++kcnt; }

    // -- stage x_t tile into LDS --
    const size_t xbase = ((size_t)m0 * T_ + t) * (size_t)G_;
    for (int i = threadIdx.x; i < 16 * G_; i += 512) {
      const int m = i >> 8, n = i & 255;
      xs[m][n] = HHin[xbase + (size_t)m * T_ * G_ + n];
    }
    if (t + 1 < T_)  // warm L2/L0 for next step's x tile
      __builtin_prefetch((const void*)(HHin + xbase + G_ + (size_t)(threadIdx.x >> 5) * T_ * G_), 0, 1);
    __syncthreads();   // xs ready; also fences last step's hs writes

    // -- Euler iterations: dh = (1-z)*(u-h) --
    for (int e = 0; e < kcnt; ++e) {
      v8f ar = {}, az = {};
      {
        const float* wr = w_hr + (size_t)nc * H_ + 2 * half;
        const float* wz = w_hz + (size_t)nc * H_ + 2 * half;
#pragma unroll 4
        for (int k = 0; k < H_; k += 4) {
          v2f a = *(const v2f*)&hs[l15][k + 2 * half];
          ar = wmma_f32_k4(a, *(const v2f*)(wr + k), ar);
          az = wmma_f32_k4(a, *(const v2f*)(wz + k), az);
        }
      }
      // rh = sigmoid(r) * h
#pragma unroll
      for (int v = 0; v < 8; ++v) {
        const int m = v + half * 8;
        rhs[m][nc] = sigm(ar[v]) * hs[m][nc];
      }
      __syncthreads();
      v8f au = {};
      {
        const float* wh = w_hh + (size_t)nc * H_ + 2 * half;
#pragma unroll 4
        for (int k = 0; k < H_; k += 4)
          au = wmma_f32_k4(*(const v2f*)&rhs[l15][k + 2 * half],
                           *(const v2f*)(wh + k), au);
      }
#pragma unroll
      for (int v = 0; v < 8; ++v) {
        const int m = v + half * 8;
        const float z = sigm(az[v]);
        const float u = tanhf(au[v]);
        const float h = hs[m][nc];
        hs[m][nc] = h + (1.0f - z) * (u - h);
      }
      __syncthreads();
    }

    // -- GRU cell: wave w computes col tile n0 of each of the 3 gates --
    v8f gir = {}, giz = {}, gin = {}, ghr = {}, ghz = {}, ghn = {};
    {
      const float* wir = w_ih  + (size_t)(0 * H_ + nc) * G_ + 2 * half;
      const float* wiz = w_ih  + (size_t)(1 * H_ + nc) * G_ + 2 * half;
      const float* win = w_ih  + (size_t)(2 * H_ + nc) * G_ + 2 * half;
      const float* whr = w_hh2 + (size_t)(0 * H_ + nc) * H_ + 2 * half;
      const float* whz = w_hh2 + (size_t)(1 * H_ + nc) * H_ + 2 * half;
      const float* whn = w_hh2 + (size_t)(2 * H_ + nc) * H_ + 2 * half;
#pragma unroll 2
      for (int k = 0; k < G_; k += 4) {
        v2f ax = *(const v2f*)&xs[l15][k + 2 * half];
        v2f ah = *(const v2f*)&hs[l15][k + 2 * half];
        gir = wmma_f32_k4(ax, *(const v2f*)(wir + k), gir);
        giz = wmma_f32_k4(ax, *(const v2f*)(wiz + k), giz);
        gin = wmma_f32_k4(ax, *(const v2f*)(win + k), gin);
        ghr = wmma_f32_k4(ah, *(const v2f*)(whr + k), ghr);
        ghz = wmma_f32_k4(ah, *(const v2f*)(whz + k), ghz);
        ghn = wmma_f32_k4(ah, *(const v2f*)(whn + k), ghn);
      }
    }
    __syncthreads();   // everyone done reading hs before we overwrite it

    {
      const float bir = b_ih[0 * H_ + nc], bhr = b_hh[0 * H_ + nc];
      const float biz = b_ih[1 * H_ + nc], bhz = b_hh[1 * H_ + nc];
      const float bin = b_ih[2 * H_ + nc], bhn = b_hh[2 * H_ + nc];
#pragma unroll
      for (int v = 0; v < 8; ++v) {
        const int m = v + half * 8;
        const float r  = sigm(gir[v] + bir + ghr[v] + bhr);
        const float z  = sigm(giz[v] + biz + ghz[v] + bhz);
        const float n  = tanhf(gin[v] + bin + r * (ghn[v] + bhn));
        const float h  = hs[m][nc];
        const float hn = (1.0f - z) * n + z * h;
        hs[m][nc] = hn;
        out[((size_t)(m0 + m) * T_ + t) * (size_t)H_ + nc] = hn;
      }
    }
    __syncthreads();
  }
}

// ---------------------------------------------------------------------------
extern "C" void kernel_launch(void* const* d_in, const int* in_sizes, int n_in,
                              void* d_out, int out_size, void* d_ws, size_t ws_size,
                              hipStream_t stream) {
  const float* Hin   = (const float*)d_in[0];
  const int*   times = (const int*)  d_in[1];
  const float* wx1   = (const float*)d_in[2];
  const float* bx1   = (const float*)d_in[3];
  const float* wx2   = (const float*)d_in[4];
  const float* bx2   = (const float*)d_in[5];
  const float* w_hr  = (const float*)d_in[6];
  const float* w_hz  = (const float*)d_in[7];
  const float* w_hh  = (const float*)d_in[8];
  const float* w_ih  = (const float*)d_in[9];
  const float* w_hh2 = (const float*)d_in[10];
  const float* b_ih  = (const float*)d_in[11];
  const float* b_hh  = (const float*)d_in[12];
  float* outp  = (float*)d_out;
  float* HHbuf = (float*)d_ws;               // [B*T, G] fp32 intermediate (128 MB)

  dim3 blk(512);
  xmodel_kernel<<<dim3((B_ * T_) / 16), blk, 0, stream>>>(Hin, wx1, bx1, wx2, bx2, HHbuf);
  scan_kernel<<<dim3(B_ / 16), blk, 0, stream>>>(HHbuf, times, w_hr, w_hz, w_hh,
                                                 w_ih, w_hh2, b_ih, b_hh, outp);
}